// MyModel_87522843561032
// MI455X (gfx1250) — compile-verified
//
#include <hip/hip_runtime.h>

// LSTM last-hidden: BATCH=16384, T=700, F=3, H=1.
// One batch element per lane; x staged through LDS in (128 batch x 20 t) tiles;
// input projection xz = x @ kernel + bias done on the matrix pipe with
// V_WMMA_F32_16X16X4_F32 (co-executes with the TRANS-heavy scan); recurrent
// part (h * recurrent_kernel, scalar since H=1) folded into the per-lane scan.

#define BATCH  16384
#define T      700
#define F      3
#define TCHUNK 20                 // 700 = 35 * 20
#define NCHUNK (T / TCHUNK)       // 35
#define BLOCK  128                // 4 wave32s per workgroup
#define NWAVES (BLOCK / 32)

typedef __attribute__((ext_vector_type(2))) float v2f;
typedef __attribute__((ext_vector_type(8))) float v8f;

__device__ __forceinline__ float fast_sigmoid(float x) {
    // 1 / (1 + e^-x), e^-x = 2^(-x*log2(e)) via v_exp_f32
    float e = __builtin_amdgcn_exp2f(-x * 1.44269504088896340736f);
    return __builtin_amdgcn_rcpf(1.0f + e);
}

__device__ __forceinline__ float fast_tanh(float x) {
    // tanh(x) = 2/(1 + e^-2x) - 1
    float e = __builtin_amdgcn_exp2f(-x * 2.88539008177792681472f);
    return 2.0f * __builtin_amdgcn_rcpf(1.0f + e) - 1.0f;
}

__global__ void lstm_scan_kernel(const float* __restrict__ x,
                                 const float* __restrict__ kern,   // (F, 4)
                                 const float* __restrict__ rkern,  // (1, 4)
                                 const float* __restrict__ bias,   // (4,)
                                 float* __restrict__ out) {        // (BATCH,)
    extern __shared__ float smem[];
    float* x_lds  = smem;                         // BLOCK * TCHUNK * 3 floats
    float* xz_lds = smem + BLOCK * TCHUNK * F;    // BLOCK * TCHUNK * 4 floats

    const int tid  = threadIdx.x;
    const int lane = tid & 31;
    const int wave = tid >> 5;
    const int b    = blockIdx.x * BLOCK + tid;    // this thread's batch element

    // ---- wave-uniform-per-lane WMMA operands (built once) ----
    // B matrix (4x16 f32, K x N): lanes 0-15 supply K=0,1; lanes 16-31 K=2,3.
    const int n  = lane & 15;                     // N column (gates live in 0..3)
    const int kb = (lane < 16) ? 0 : 2;           // base K row for this half-wave
    v2f bmat;
    bmat.x = (n < 4 && (kb + 0) < F) ? kern[(kb + 0) * 4 + n] : 0.0f;
    bmat.y = (n < 4 && (kb + 1) < F) ? kern[(kb + 1) * 4 + n] : 0.0f;

    // C matrix: bias broadcast along rows -> every VGPR = bias[N].
    const float bias_n = (n < 4) ? bias[n] : 0.0f;
    v8f cmat;
#pragma unroll
    for (int v = 0; v < 8; ++v) cmat[v] = bias_n;

    // A.y mask: upper half-wave supplies K=3, which is the zero-padded row.
    // Load unconditionally (the stray read lands inside our LDS allocation)
    // and multiply the value away -- avoids a predicated DS load + exec dance.
    const float aymask = (kb == 0) ? 1.0f : 0.0f;

    const float rk0 = rkern[0], rk1 = rkern[1], rk2 = rkern[2], rk3 = rkern[3];

    float h = 0.0f, c = 0.0f;
    const float* xg = x + (size_t)b * (T * F);

    for (int chunk = 0; chunk < NCHUNK; ++chunk) {
        const int t0 = chunk * TCHUNK;

        __syncthreads();  // previous chunk's readers done before we overwrite LDS

        // ---- Phase A: stage x tile. Each thread copies its own contiguous
        // TCHUNK*3 = 60 floats (15 aligned float4 loads). ----
        {
            const float4* src = (const float4*)(xg + t0 * F);
            float4*       dst = (float4*)(x_lds + tid * (TCHUNK * F));
#pragma unroll
            for (int v = 0; v < (TCHUNK * F) / 4; ++v) dst[v] = src[v];
        }
        __syncthreads();

        // ---- Phase B: xz = x @ kernel + bias on the matrix pipe.
        // Rows r = b_local*TCHUNK + t_local; each wave owns 2*TCHUNK groups
        // of 16 rows. A (16x4 f32): lane m holds (K0,K1)=(x0,x1), lane 16+m
        // holds (K2,K3)=(x2,0). ----
        {
            const int rbase_wave = wave * (2 * TCHUNK * 16);
            for (int g = 0; g < 2 * TCHUNK; ++g) {
                const int rb = rbase_wave + g * 16;
                const int r  = rb + n;
                v2f a;
                a.x = x_lds[r * F + kb];
                a.y = x_lds[r * F + kb + 1] * aymask;

                v8f d = __builtin_amdgcn_wmma_f32_16x16x4_f32(
                    false, a, false, bmat, (short)0, cmat, false, false);

                // Scatter D: lane's column N=n (gates 0..3 only); VGPR v is
                // row v (lanes 0-15) or row 8+v (lanes 16-31).
                if (n < 4) {
                    const int mofs = (lane < 16) ? 0 : 8;
#pragma unroll
                    for (int v = 0; v < 8; ++v)
                        xz_lds[(rb + mofs + v) * 4 + n] = d[v];
                }
            }
        }
        __syncthreads();

        // ---- Phase C: sequential LSTM scan, one batch element per lane.
        // z = xz + h * recurrent_kernel; gate order i, f, g, o. ----
        {
            const float4* zz = (const float4*)(xz_lds + tid * (TCHUNK * 4));
#pragma unroll
            for (int t = 0; t < TCHUNK; ++t) {
                float4 z = zz[t];
                float ig = fast_sigmoid(z.x + h * rk0);
                float fg = fast_sigmoid(z.y + h * rk1);
                float gg = fast_tanh   (z.z + h * rk2);
                float og = fast_sigmoid(z.w + h * rk3);
                c = fg * c + ig * gg;
                h = og * fast_tanh(c);
            }
        }
    }

    out[b] = h;
}

extern "C" void kernel_launch(void* const* d_in, const int* in_sizes, int n_in,
                              void* d_out, int out_size, void* d_ws, size_t ws_size,
                              hipStream_t stream) {
    const float* x  = (const float*)d_in[0];
    const float* k  = (const float*)d_in[1];
    const float* rk = (const float*)d_in[2];
    const float* bi = (const float*)d_in[3];
    float*       o  = (float*)d_out;

    const size_t lds_bytes = (size_t)BLOCK * TCHUNK * (F + 4) * sizeof(float); // 70 KB
    hipLaunchKernelGGL(lstm_scan_kernel,
                       dim3(BATCH / BLOCK), dim3(BLOCK), lds_bytes, stream,
                       x, k, rk, bi, o);
}